// NoisyRNN_30880814858792
// MI455X (gfx1250) — compile-verified
//
#include <hip/hip_runtime.h>

typedef __attribute__((ext_vector_type(2))) float v2f;
typedef __attribute__((ext_vector_type(4))) float v4f;
typedef __attribute__((ext_vector_type(8))) float v8f;

#define N_UNITS 128
#define D_IN    64
#define T_STEPS 1024
#define BM      16          // batch rows per workgroup
#define HS      132         // padded LDS row stride for h (floats) -> bank-friendly
#define XS      68          // padded LDS row stride for x tile (floats)
#define EPS_C     0.01f
#define GAMMA_A_C 0.01f
#define GAMMA_W_C 0.01f
// BETA = 0.8:  BETA*(X - X^T) + (1-BETA)*(X + X^T) = X + (1-2*BETA)*X^T = X - 0.6*X^T

__global__ __launch_bounds__(256, 1)
void noisy_rnn_kernel(const float* __restrict__ x,
                      const float* __restrict__ E_w,
                      const float* __restrict__ E_b,
                      const float* __restrict__ B_p,
                      const float* __restrict__ C_p,
                      const float* __restrict__ D_w,
                      const float* __restrict__ D_b,
                      float* __restrict__ out)
{
    __shared__ float h_buf[2][BM * HS];   // double-buffered h tile (16 x 128, padded)
    __shared__ float x_buf[2][BM * XS];   // double-buffered x_t tile (16 x 64, padded)

    const int tid  = threadIdx.x;
    const int lane = tid & 31;
    const int wave = tid >> 5;            // 0..7 : which 16-column tile of the 128 units
    const int col0 = wave * 16;
    const int b0   = blockIdx.x * BM;

    const int lrow = lane & 15;           // M row (A-frag) / N col (B-frag)
    const int lhi  = lane >> 4;           // 0 or 1 (K sub-select per WMMA layout)

    // ---------------- Persistent register-resident B-operand fragments ----------------
    // B-frag layout for 4x16 f32 B operand: .x = B[4c + lhi][n], .y = B[4c + 2 + lhi][n]
    v2f fragA[32], fragW[32], fragE[16];
    {
        const int n = col0 + lrow;
        #pragma unroll
        for (int c = 0; c < 32; ++c) {
            const int k0 = 4 * c + lhi;
            const int k1 = k0 + 2;
            fragA[c].x = B_p[k0 * 128 + n] - 0.6f * B_p[n * 128 + k0] - (k0 == n ? GAMMA_A_C : 0.0f);
            fragA[c].y = B_p[k1 * 128 + n] - 0.6f * B_p[n * 128 + k1] - (k1 == n ? GAMMA_A_C : 0.0f);
            fragW[c].x = C_p[k0 * 128 + n] - 0.6f * C_p[n * 128 + k0] - (k0 == n ? GAMMA_W_C : 0.0f);
            fragW[c].y = C_p[k1 * 128 + n] - 0.6f * C_p[n * 128 + k1] - (k1 == n ? GAMMA_W_C : 0.0f);
        }
        #pragma unroll
        for (int c = 0; c < 16; ++c) {
            const int k0 = 4 * c + lhi;          // B[k][n] = E_w[n][k]  (z = x @ E_w^T)
            fragE[c].x = E_w[n * D_IN + k0];
            fragE[c].y = E_w[n * D_IN + k0 + 2];
        }
    }
    const float eb = E_b[col0 + lrow];

    // ---------------- init: h = 0, preload x_0 tile ----------------
    for (int i = tid; i < BM * HS; i += 256) h_buf[0][i] = 0.0f;
    {
        const int r = tid >> 4, i4 = (tid & 15) * 4;
        const v4f v = *(const v4f*)(x + ((size_t)(b0 + r) * T_STEPS + 0) * D_IN + i4);
        float* dst = &x_buf[0][r * XS + i4];
        dst[0] = v.x; dst[1] = v.y; dst[2] = v.z; dst[3] = v.w;
    }
    v8f h_d = {};   // this wave's 16x16 slice of h, C/D register layout
    __syncthreads();

    // ---------------- sequential recurrence ----------------
    for (int t = 0; t < T_STEPS; ++t) {
        const int cur = t & 1, nxt = cur ^ 1;

        // prefetch x_{t+1} (overlaps with WMMA work below)
        v4f xv = {};
        const int pr = tid >> 4, pi4 = (tid & 15) * 4;
        if (t + 1 < T_STEPS)
            xv = *(const v4f*)(x + ((size_t)(b0 + pr) * T_STEPS + (t + 1)) * D_IN + pi4);

        v8f accW;                       // will hold h@W + x@E_w^T + E_b
        #pragma unroll
        for (int j = 0; j < 8; ++j) accW[j] = eb;
        v8f accA = {};                  // will hold h@A

        // z contribution: x_t (16x64) @ E_w^T (64x16-slice), K=64 -> 16 WMMAs
        const float* xb = x_buf[cur];
        #pragma unroll
        for (int c = 0; c < 16; ++c) {
            const v2f af = *(const v2f*)&xb[lrow * XS + 4 * c + 2 * lhi];
            accW = __builtin_amdgcn_wmma_f32_16x16x4_f32(false, af, false, fragE[c],
                                                         (short)0, accW, false, false);
        }
        // recurrent matmuls: h (16x128) @ A and @ W (128x16 slices), K=128 -> 2x32 WMMAs
        const float* hb = h_buf[cur];
        #pragma unroll
        for (int c = 0; c < 32; ++c) {
            const v2f hf = *(const v2f*)&hb[lrow * HS + 4 * c + 2 * lhi];
            accA = __builtin_amdgcn_wmma_f32_16x16x4_f32(false, hf, false, fragA[c],
                                                         (short)0, accA, false, false);
            accW = __builtin_amdgcn_wmma_f32_16x16x4_f32(false, hf, false, fragW[c],
                                                         (short)0, accW, false, false);
        }

        // h += EPS * (ALPHA * h@A + tanh(h@W + z))   (ALPHA = 1)
        #pragma unroll
        for (int j = 0; j < 8; ++j)
            h_d[j] = h_d[j] + EPS_C * (accA[j] + tanhf(accW[j]));

        // publish new h slice (C/D layout: row M = j + 8*lhi, col N = col0 + lrow)
        float* hn = h_buf[nxt];
        #pragma unroll
        for (int j = 0; j < 8; ++j)
            hn[(j + 8 * lhi) * HS + col0 + lrow] = h_d[j];

        // publish prefetched x_{t+1}
        if (t + 1 < T_STEPS) {
            float* dst = &x_buf[nxt][pr * XS + pi4];
            dst[0] = xv.x; dst[1] = xv.y; dst[2] = xv.z; dst[3] = xv.w;
        }
        __syncthreads();
    }

    // ---------------- output head: out = h_final @ D_w^T + D_b  (16 x 10 per WG) ----------------
    const float* hfin = h_buf[T_STEPS & 1];
    if (tid < BM * 10) {
        const int r = tid / 10, c = tid % 10;
        float s = D_b[c];
        #pragma unroll 8
        for (int u = 0; u < N_UNITS; ++u)
            s += hfin[r * HS + u] * D_w[c * N_UNITS + u];
        out[(b0 + r) * 10 + c] = s;
    }
}

extern "C" void kernel_launch(void* const* d_in, const int* in_sizes, int n_in,
                              void* d_out, int out_size, void* d_ws, size_t ws_size,
                              hipStream_t stream) {
    const float* x   = (const float*)d_in[0];
    const float* E_w = (const float*)d_in[1];
    const float* E_b = (const float*)d_in[2];
    const float* B_p = (const float*)d_in[3];
    const float* C_p = (const float*)d_in[4];
    const float* D_w = (const float*)d_in[5];
    const float* D_b = (const float*)d_in[6];
    float* out = (float*)d_out;

    dim3 grid(512 / BM);   // 32 workgroups, each owns 16 batch rows
    dim3 block(256);       // 8 waves of 32
    noisy_rnn_kernel<<<grid, block, 0, stream>>>(x, E_w, E_b, B_p, C_p, D_w, D_b, out);
}